// ACE_15479062135011
// MI455X (gfx1250) — compile-verified
//
#include <hip/hip_runtime.h>
#include <hip/hip_bf16.h>

// Shapes from the reference
#define T_DIM 512
#define B_DIM 32
#define V_DIM 10000
#define L_DIM 64

typedef __attribute__((ext_vector_type(2))) float v2f;
typedef __attribute__((ext_vector_type(8))) float v8f;

// ---------------------------------------------------------------------------
// Kernel 1: T-reduction of x[T,B,V] -> sums[B,V] using V_WMMA_F32_16X16X4_F32.
//
// Per wave: one batch b, one 16-wide v tile. A = ones(16x4), B = x tile
// (K = 4 t-steps, N = 16 v columns), C accumulates in f32:
//   D[m,n] = sum_k B[k,n] + C[m,n]  -> every row m holds the running colsum.
// B-matrix f32 4x16 VGPR layout (ISA 7.12.2, "row striped across lanes"):
//   lanes 0-15 : VGPR0 = K0 row, VGPR1 = K1 row  (N = lane)
//   lanes 16-31: VGPR0 = K2 row, VGPR1 = K3 row  (N = lane-16)
// Since A is all-ones the sum is invariant to the K-slot assignment; only
// the lane<->N mapping must match D's layout (it does: N = lane&15).
// ---------------------------------------------------------------------------
__global__ void __launch_bounds__(256)
ctc_colsum_wmma(const float* __restrict__ x, float* __restrict__ sums) {
    const int b    = blockIdx.x;                      // batch
    const int wave = threadIdx.x >> 5;                // 8 waves per block
    const int tile = blockIdx.y * 8 + wave;           // 16-wide v tile index
    if (tile >= V_DIM / 16) return;                   // uniform per wave -> EXEC stays full

    const int lane = threadIdx.x & 31;
    const int vcol = lane & 15;                       // N within tile
    const int thalf = (lane >> 4) << 1;               // t-offset 0 or 2 for this lane group
    const size_t BV = (size_t)B_DIM * V_DIM;          // t-stride in elements

    // This lane's base element: x[thalf, b, tile*16 + vcol]
    const float* p = x + (size_t)thalf * BV + (size_t)b * V_DIM + (size_t)tile * 16 + vcol;

    v2f a; a.x = 1.0f; a.y = 1.0f;                    // ones A-matrix (layout-independent)
    v8f c = {0.f, 0.f, 0.f, 0.f, 0.f, 0.f, 0.f, 0.f};

#pragma unroll 8
    for (int t = 0; t < T_DIM; t += 4) {
        v2f bm;
        bm.x = p[(size_t)t * BV];                     // K slot 0 (or 2) for this lane group
        bm.y = p[(size_t)t * BV + BV];                // K slot 1 (or 3)
        // 8 args: (neg_a, A, neg_b, B, c_mod, C, reuse_a, reuse_b)
        c = __builtin_amdgcn_wmma_f32_16x16x4_f32(false, a, false, bm,
                                                  (short)0, c, false, false);
    }

    // Row M=0 of D: VGPR0, lanes 0-15, N = lane. (Lanes 16-31 hold row M=8,
    // identical values; only lanes 0-15 store.)
    if (lane < 16) {
        sums[(size_t)b * V_DIM + (size_t)tile * 16 + lane] = c[0];
    }
}

// ---------------------------------------------------------------------------
// Kernel 2: epilogue. counts[b,v] is nonzero only for v in {labels[b]} u {0},
// so the loss needs only <=65 log() terms per batch:
//   loss = -(1/(B*T)) * sum_b [ sum_{l: lab!=0} log(xavg[b,lab])
//                               + (T - len_b) * log(xavg[b,0]) ]
// with xavg = sum/T + 1e-10. Single block -> deterministic single write.
// ---------------------------------------------------------------------------
__global__ void __launch_bounds__(256)
ctc_loss_epilogue(const float* __restrict__ sums, const int* __restrict__ label,
                  float* __restrict__ out) {
    __shared__ float partial[256];
    __shared__ int   lens[B_DIM];

    const int tid = threadIdx.x;
    if (tid < B_DIM) lens[tid] = 0;
    __syncthreads();

    const float invT = 1.0f / (float)T_DIM;
    float acc = 0.0f;

    for (int i = tid; i < B_DIM * L_DIM; i += 256) {
        const int b = i >> 6;            // /64
        const int v = label[i];
        if (v != 0) {
            atomicAdd(&lens[b], 1);
            const float xa = sums[(size_t)b * V_DIM + v] * invT + 1e-10f;
            acc += logf(xa);
        }
    }
    partial[tid] = acc;
    __syncthreads();                      // lens finalized here

    if (tid < B_DIM) {
        const float xa0 = sums[(size_t)tid * V_DIM] * invT + 1e-10f;
        partial[tid] += ((float)T_DIM - (float)lens[tid]) * logf(xa0);
    }
    __syncthreads();

    for (int s = 128; s > 0; s >>= 1) {
        if (tid < s) partial[tid] += partial[tid + s];
        __syncthreads();
    }
    if (tid == 0) {
        out[0] = -partial[0] / ((float)B_DIM * (float)T_DIM);
    }
}

extern "C" void kernel_launch(void* const* d_in, const int* in_sizes, int n_in,
                              void* d_out, int out_size, void* d_ws, size_t ws_size,
                              hipStream_t stream) {
    const float* x     = (const float*)d_in[0];   // [T, B, V] f32
    const int*   label = (const int*)d_in[1];     // [B, L] int
    float*       out   = (float*)d_out;           // scalar f32
    float*       sums  = (float*)d_ws;            // [B, V] f32 partial sums (1.28 MB)

    // 625 tiles of 16 v per batch; 8 waves/block -> ceil(625/8) = 79 blocks in y.
    dim3 grid(B_DIM, 79, 1);
    ctc_colsum_wmma<<<grid, 256, 0, stream>>>(x, sums);
    ctc_loss_epilogue<<<1, 256, 0, stream>>>(sums, label, out);
}